// BatchedExponentialOrthogonalization_19928648254160
// MI455X (gfx1250) — compile-verified
//
#include <hip/hip_runtime.h>

typedef __attribute__((ext_vector_type(2))) float v2f;
typedef __attribute__((ext_vector_type(4))) float v4f;
typedef __attribute__((ext_vector_type(8))) float v8f;
typedef __attribute__((ext_vector_type(4))) int   v4i;

typedef v4i __attribute__((address_space(1)))* global_v4i_ptr;
typedef v4i __attribute__((address_space(3)))* lds_v4i_ptr;

#define MAT 256
#define NBATCH 512
#define NITERS 7
#define MTILES 2
#define NTILES 16
#define ROWS_PER_WAVE 32

// ---------------- async global->LDS helpers (guarded; always compiles) ----------------
__device__ __forceinline__ void async_copy_b128(const float* g, float* l) {
#if __has_builtin(__builtin_amdgcn_global_load_async_to_lds_b128)
  __builtin_amdgcn_global_load_async_to_lds_b128(
      (global_v4i_ptr)(v4i*)g, (lds_v4i_ptr)(v4i*)l, 0, 0);
#elif __has_builtin(__builtin_amdgcn_load_to_lds)
  __builtin_amdgcn_load_to_lds((void*)g,
      (__attribute__((address_space(3))) void*)l, 16, 0, 0);
#else
  *(v4f*)l = *(const v4f*)g;
#endif
}

__device__ __forceinline__ void wait_async0() {
#if __has_builtin(__builtin_amdgcn_s_wait_asynccnt)
  __builtin_amdgcn_s_wait_asynccnt(0);
#else
  asm volatile("s_wait_asynccnt 0x0" ::: "memory");
#endif
}

// =====================================================================================
// Kernel 1: pack S = 0.5*(W - W^T) into ws in B-fragment-friendly packed-pair layout
//           ws[b][k2][n*2 + (k&1)] = S[k][n]  (k2 = k>>1, row stride 512 floats)
//           and write res = I + S to d_out.
// =====================================================================================
#define P1_K 16
#define P1_STRIDE 520   // LDS pad: 520 % 64 == 8 -> conflict-light transpose RMW

__global__ __launch_bounds__(256)
void pack_skew_kernel(const float* __restrict__ w, float* __restrict__ wsS,
                      float* __restrict__ out) {
  __shared__ float sL[(P1_K / 2) * P1_STRIDE];   // 16,640 B
  const int b   = blockIdx.x >> 4;
  const int p   = blockIdx.x & 15;               // 16 panels of 16 rows
  const int tid = threadIdx.x;
  const float* wb  = w   + (size_t)b * MAT * MAT;
  float*       wsb = wsS + (size_t)b * MAT * MAT;
  float*       ob  = out + (size_t)b * MAT * MAT;

  #pragma unroll 4
  for (int f = tid; f < P1_K * MAT; f += 256) {  // direct rows (coalesced over n)
    int n = f & (MAT - 1), k = f >> 8, kg = p * P1_K + k;
    sL[(k >> 1) * P1_STRIDE + n * 2 + (k & 1)] = 0.5f * wb[kg * MAT + n];
  }
  __syncthreads();
  #pragma unroll 4
  for (int f = tid; f < P1_K * MAT; f += 256) {  // transpose (coalesced over k)
    int k = f & (P1_K - 1), n = f >> 4, kg = p * P1_K + k;
    sL[(k >> 1) * P1_STRIDE + n * 2 + (k & 1)] -= 0.5f * wb[n * MAT + kg];
  }
  __syncthreads();
  #pragma unroll 4
  for (int f = tid; f < P1_K * MAT; f += 256) {  // packed S -> ws (coalesced)
    int k2 = f >> 9, col = f & 511;
    wsb[(size_t)(p * (P1_K / 2) + k2) * 512 + col] = sL[k2 * P1_STRIDE + col];
  }
  #pragma unroll 4
  for (int f = tid; f < P1_K * MAT; f += 256) {  // res = I + S (coalesced)
    int n = f & (MAT - 1), k = f >> 8, kg = p * P1_K + k;
    float v = sL[(k >> 1) * P1_STRIDE + n * 2 + (k & 1)];
    ob[kg * MAT + n] = v + ((kg == n) ? 1.0f : 0.0f);
  }
}

// =====================================================================================
// Kernel 2: series acc_i = acc_{i-1} @ S / i ; res += acc_i.
//   accL (acc_{i-1}, row-major, padded) lives in LDS; S panels stream in via
//   double-buffered async global->LDS copies overlapped with the WMMA block.
// =====================================================================================
#define ACC_STRIDE 260          // 260 % 64 == 4 -> conflict-free b64 A-frag loads
#define P2_K 16                 // K-panel rows
#define NP2 (MAT / P2_K)        // 16 panels
#define K2S 516                 // LDS panel stride (floats); 516*4 = 2064 B, 16B aligned
#define PANEL_FLOATS ((P2_K / 2) * K2S)
#define SH2_FLOATS (MAT * ACC_STRIDE + 2 * PANEL_FLOATS)   // 74,816 f = 299,264 B

__device__ __forceinline__ void copy_panel_async(const float* __restrict__ wsb,
                                                 float* __restrict__ dst,
                                                 int p, int tid) {
  // panel = 8 packed k2-rows x 512 floats = 16 KB = 1024 b128 chunks
  #pragma unroll
  for (int j = 0; j < 4; ++j) {
    int q = tid + 256 * j;
    int k2l = q >> 7;                 // 0..7
    int c4  = q & 127;                // 0..127
    async_copy_b128(wsb + (size_t)(p * (P2_K / 2) + k2l) * 512 + c4 * 4,
                    dst + k2l * K2S + c4 * 4);
  }
}

__global__ __launch_bounds__(256)
void series_kernel(const float* __restrict__ wsS, float* __restrict__ out) {
  extern __shared__ float lds[];
  float* accL  = lds;                             // [MAT][ACC_STRIDE]
  float* sbase = lds + MAT * ACC_STRIDE;          // two panel buffers follow

  const int b    = blockIdx.x;
  const int tid  = threadIdx.x;
  const int lane = tid & 31;
  const int wave = tid >> 5;
  const float* wsb = wsS + (size_t)b * MAT * MAT;
  float*       ob  = out + (size_t)b * MAT * MAT;
  const int mbase = wave * ROWS_PER_WAVE;

  // acc_1 = S : unpack packed ws into row-major accL (coalesced global reads)
  #pragma unroll 4
  for (int f = tid; f < MAT * MAT; f += 256) {
    int k2 = f >> 9, col = f & 511;
    int k = 2 * k2 + (col & 1), n = col >> 1;
    accL[k * ACC_STRIDE + n] = wsb[f];
  }

  // prologue: panel 0 -> buf0
  copy_panel_async(wsb, sbase, 0, tid);
  wait_async0();
  __syncthreads();

  int curbuf = 0;
  for (int it = 2; it < NITERS; ++it) {
    const float inv_i = 1.0f / (float)it;

    v8f c[MTILES][NTILES];
    #pragma unroll
    for (int m = 0; m < MTILES; ++m)
      #pragma unroll
      for (int n = 0; n < NTILES; ++n) { v8f z{}; c[m][n] = z; }

    #pragma unroll 1
    for (int p = 0; p < NP2; ++p) {
      // prefetch next panel into the other buffer (overlaps the WMMA block)
      const bool last_step = (it == NITERS - 1) && (p == NP2 - 1);
      if (!last_step)
        copy_panel_async(wsb, sbase + (curbuf ^ 1) * PANEL_FLOATS,
                         (p + 1) & (NP2 - 1), tid);

      const float* sP = sbase + curbuf * PANEL_FLOATS;
      #pragma unroll
      for (int k4 = 0; k4 < P2_K / 4; ++k4) {
        const int bbase = (k4 * 2 + (lane >> 4)) * K2S + (lane & 15) * 2;
        v2f bfr[NTILES];
        #pragma unroll
        for (int n = 0; n < NTILES; ++n)
          bfr[n] = *(const v2f*)(sP + bbase + n * 32);

        #pragma unroll
        for (int m = 0; m < MTILES; ++m) {
          const int row  = mbase + m * 16 + (lane & 15);
          const int koff = p * P2_K + k4 * 4 + ((lane >> 4) << 1);
          v2f afr = *(const v2f*)(accL + row * ACC_STRIDE + koff);
          #pragma unroll
          for (int n = 0; n < NTILES; ++n)
            c[m][n] = __builtin_amdgcn_wmma_f32_16x16x4_f32(
                false, afr, false, bfr[n], (short)0, c[m][n], false, false);
        }
      }

      wait_async0();        // only this panel-body's copies are outstanding
      __syncthreads();      // everyone's copies visible; everyone done reading sP
      curbuf ^= 1;
    }

    // finalize: acc_i = C/i -> accL (own rows only); res += acc_i (fire-and-forget)
    #pragma unroll
    for (int m = 0; m < MTILES; ++m) {
      const int row0 = mbase + m * 16 + ((lane >> 4) << 3);
      const int colb = lane & 15;
      #pragma unroll
      for (int n = 0; n < NTILES; ++n) {
        const int col = n * 16 + colb;
        #pragma unroll
        for (int v = 0; v < 8; ++v) {
          float val = c[m][n][v] * inv_i;
          const int r = row0 + v;
          accL[r * ACC_STRIDE + col] = val;
          unsafeAtomicAdd(ob + r * MAT + col, val);   // global_atomic_add_f32, no return
        }
      }
    }
    // accL rows are wave-private; sP hazards handled by in-loop wait+barrier.
  }
}

// =====================================================================================
// Fallback (ws too small): round-1 monolithic kernel — recomputes S panels from w.
// =====================================================================================
#define M_PANEL_K 32
#define M_NPANELS (MAT / M_PANEL_K)
#define M_K2S 520
#define SHM_FLOATS (MAT * ACC_STRIDE + (M_PANEL_K / 2) * M_K2S)

__device__ __forceinline__ void fill_skew_panel(const float* __restrict__ wb,
                                                float* __restrict__ sP,
                                                int p, int tid) {
  #pragma unroll 4
  for (int f = tid; f < M_PANEL_K * MAT; f += 256) {
    int n = f & (MAT - 1), k = f >> 8, kg = p * M_PANEL_K + k;
    sP[(k >> 1) * M_K2S + n * 2 + (k & 1)] = 0.5f * wb[kg * MAT + n];
  }
  __syncthreads();
  #pragma unroll 4
  for (int f = tid; f < M_PANEL_K * MAT; f += 256) {
    int k = f & (M_PANEL_K - 1), n = f >> 5, kg = p * M_PANEL_K + k;
    sP[(k >> 1) * M_K2S + n * 2 + (k & 1)] -= 0.5f * wb[n * MAT + kg];
  }
}

__global__ __launch_bounds__(256)
void skew_exp_series_mono(const float* __restrict__ w, float* __restrict__ out) {
  extern __shared__ float lds[];
  float* accL = lds;
  float* sP   = lds + MAT * ACC_STRIDE;
  const int b = blockIdx.x, tid = threadIdx.x, lane = tid & 31, wave = tid >> 5;
  const float* wb = w + (size_t)b * MAT * MAT;
  float*       ob = out + (size_t)b * MAT * MAT;

  for (int p = 0; p < M_NPANELS; ++p) {
    __syncthreads();
    fill_skew_panel(wb, sP, p, tid);
    __syncthreads();
    #pragma unroll 4
    for (int f = tid; f < M_PANEL_K * MAT; f += 256) {
      int n = f & (MAT - 1), k = f >> 8, kg = p * M_PANEL_K + k;
      float v = sP[(k >> 1) * M_K2S + n * 2 + (k & 1)];
      ob[kg * MAT + n]          = v + ((kg == n) ? 1.0f : 0.0f);
      accL[kg * ACC_STRIDE + n] = v;
    }
  }
  const int mbase = wave * ROWS_PER_WAVE;

  for (int it = 2; it < NITERS; ++it) {
    const float inv_i = 1.0f / (float)it;
    v8f c[MTILES][NTILES];
    #pragma unroll
    for (int m = 0; m < MTILES; ++m)
      #pragma unroll
      for (int n = 0; n < NTILES; ++n) { v8f z{}; c[m][n] = z; }

    #pragma unroll 1
    for (int p = 0; p < M_NPANELS; ++p) {
      __syncthreads();
      fill_skew_panel(wb, sP, p, tid);
      __syncthreads();
      #pragma unroll
      for (int k4 = 0; k4 < M_PANEL_K / 4; ++k4) {
        const int bbase = (k4 * 2 + (lane >> 4)) * M_K2S + (lane & 15) * 2;
        v2f bfr[NTILES];
        #pragma unroll
        for (int n = 0; n < NTILES; ++n)
          bfr[n] = *(const v2f*)(sP + bbase + n * 32);
        #pragma unroll
        for (int m = 0; m < MTILES; ++m) {
          const int row  = mbase + m * 16 + (lane & 15);
          const int koff = p * M_PANEL_K + k4 * 4 + ((lane >> 4) << 1);
          v2f afr = *(const v2f*)(accL + row * ACC_STRIDE + koff);
          #pragma unroll
          for (int n = 0; n < NTILES; ++n)
            c[m][n] = __builtin_amdgcn_wmma_f32_16x16x4_f32(
                false, afr, false, bfr[n], (short)0, c[m][n], false, false);
        }
      }
    }
    #pragma unroll
    for (int m = 0; m < MTILES; ++m) {
      const int row0 = mbase + m * 16 + ((lane >> 4) << 3);
      const int colb = lane & 15;
      #pragma unroll
      for (int n = 0; n < NTILES; ++n) {
        const int col = n * 16 + colb;
        #pragma unroll
        for (int v = 0; v < 8; ++v) {
          float val = c[m][n][v] * inv_i;
          const int r = row0 + v;
          ob[r * MAT + col] += val;
          accL[r * ACC_STRIDE + col] = val;
        }
      }
    }
  }
}

// =====================================================================================
extern "C" void kernel_launch(void* const* d_in, const int* in_sizes, int n_in,
                              void* d_out, int out_size, void* d_ws, size_t ws_size,
                              hipStream_t stream) {
  const float* w = (const float*)d_in[0];
  float* out = (float*)d_out;
  (void)in_sizes; (void)n_in; (void)out_size;

  const size_t needS = (size_t)NBATCH * MAT * MAT * sizeof(float);  // 128 MB
  if (ws_size >= needS && d_ws != nullptr) {
    float* wsS = (float*)d_ws;
    pack_skew_kernel<<<dim3(NBATCH * 16), dim3(256), 0, stream>>>(w, wsS, out);
    series_kernel<<<dim3(NBATCH), dim3(256),
                    (size_t)SH2_FLOATS * sizeof(float), stream>>>(wsS, out);
  } else {
    skew_exp_series_mono<<<dim3(NBATCH), dim3(256),
                           (size_t)SHM_FLOATS * sizeof(float), stream>>>(w, out);
  }
}